// MarsagliaTsampler_18734647345742
// MI455X (gfx1250) — compile-verified
//
#include <hip/hip_runtime.h>
#include <stdint.h>

#define TOPK  10
#define CHUNK 16   // rows staged per async round (32 async loads in flight, ASYNCcnt max 63)

// ---- CDNA5 async-tensor path helpers (inline asm; portable across toolchains) ----

// LDS byte offset of a generic pointer to a __shared__ object.
__device__ __forceinline__ uint32_t lds_addr_u32(const void* p) {
  return (uint32_t)(size_t)(__attribute__((address_space(3))) const void*)p;
}

// GLOBAL_LOAD_ASYNC_TO_LDS_B32: per-lane 4B copy, memory -> LDS, tracked by ASYNCcnt.
__device__ __forceinline__ void async_load_b32(uint32_t lds_dst, const float* gsrc) {
  asm volatile("global_load_async_to_lds_b32 %0, %1, off"
               :
               : "v"(lds_dst), "v"((uint64_t)(uintptr_t)gsrc)
               : "memory");
}

__device__ __forceinline__ void wait_asynccnt0() {
  asm volatile("s_wait_asynccnt 0" ::: "memory");
}

// One wave (32 lanes) per 32 consecutive columns. Scan rows in order,
// emit first TOPK accepted values per column, zero-fill the remainder.
__global__ __launch_bounds__(32) void marsaglia_topk_kernel(
    const float* __restrict__ Z, const float* __restrict__ U,
    const float* __restrict__ gamma_alpha, float* __restrict__ out,
    int B, int S)
{
  __shared__ float zbuf[CHUNK * 32];
  __shared__ float ubuf[CHUNK * 32];

  const int  lane   = (int)threadIdx.x;
  const int  col    = (int)blockIdx.x * 32 + lane;
  const bool active = (col < S);
  const int  acol   = active ? col : (S - 1);   // clamp so inactive lanes read in-bounds

  // Per-column constants: alpha = relu(ga)+1, d = alpha-1/3, c = 1/sqrt(9d)
  const float ga    = gamma_alpha[acol];
  const float alpha = fmaxf(ga, 0.0f) + 1.0f;
  const float d     = alpha - (1.0f / 3.0f);
  const float c     = 1.0f / sqrtf(9.0f * d);
  const float zmin  = -1.0f / c;                // cond1: Z > -1/c

  if (active) out[TOPK * S + col] = alpha;      // second tuple element

  const uint32_t zb = lds_addr_u32(&zbuf[lane]);
  const uint32_t ub = lds_addr_u32(&ubuf[lane]);

  int count = active ? 0 : TOPK;                // inactive lanes never block early exit

  for (int row0 = 0; row0 < B; row0 += CHUNK) {
    int rows = B - row0; if (rows > CHUNK) rows = CHUNK;

    // Stage a (rows x 32) tile of Z and U into LDS via the async data mover.
    const float* zp = Z + (size_t)row0 * (size_t)S + (size_t)acol;
    const float* up = U + (size_t)row0 * (size_t)S + (size_t)acol;
    for (int r = 0; r < rows; ++r) {
      async_load_b32(zb + (uint32_t)(r * 32 * 4), zp + (size_t)r * (size_t)S);
      async_load_b32(ub + (uint32_t)(r * 32 * 4), up + (size_t)r * (size_t)S);
    }
    wait_asynccnt0();

    // Consume the tile: rows in order, so "first TOPK positives" is preserved.
    for (int r = 0; r < rows; ++r) {
      const float z = zbuf[r * 32 + lane];
      const float u = ubuf[r * 32 + lane];
      const float t = 1.0f + c * z;
      const float V = t * t * t;
      // NaN from logf(V<=0) makes the comparison false, matching JAX semantics.
      const float rhs = 0.5f * (z * z) + d - d * V + d * logf(V);
      const bool  acc = (z > zmin) && (logf(u) < rhs);
      if (acc && (count < TOPK)) {
        out[count * S + col] = d * V;           // out = cond*d*V, cond==1 here
        ++count;
      }
    }

    // Early exit once every column owned by this wave has its TOPK values
    // (~16-32 rows in practice -> we read ~0.3% of the 256 MB input).
    if (__ballot(count < TOPK) == 0) break;
  }

  // Pad: reference gathers rows with out<=0, whose values are exactly 0.
  if (active) {
    for (int k = count; k < TOPK; ++k) out[k * S + col] = 0.0f;
  }
}

extern "C" void kernel_launch(void* const* d_in, const int* in_sizes, int n_in,
                              void* d_out, int out_size, void* d_ws, size_t ws_size,
                              hipStream_t stream) {
  (void)n_in; (void)d_ws; (void)ws_size; (void)out_size;
  const float* Z  = (const float*)d_in[0];
  const float* U  = (const float*)d_in[1];
  const float* ga = (const float*)d_in[2];
  // batch_size (d_in[3]) lives on device; derive shapes from in_sizes instead.
  const int S = in_sizes[2];
  const int B = in_sizes[0] / S;

  float* out = (float*)d_out;
  const int nblocks = (S + 31) / 32;
  marsaglia_topk_kernel<<<nblocks, 32, 0, stream>>>(Z, U, ga, out, B, S);
}